// NLGNN2_15075335209142
// MI455X (gfx1250) — compile-verified
//
#include <hip/hip_runtime.h>
#include <cstddef>
#include <cstdint>

typedef __attribute__((ext_vector_type(2))) float v2f;
typedef __attribute__((ext_vector_type(8))) float v8f;

// ---------------------------------------------------------------------------
// WMMA GEMM: C[M x 64] = A[M x K] @ B[K x 64] (+bias)(+relu)
// One wave (32 lanes) computes a 16x16 tile with V_WMMA_F32_16X16X4_F32.
// Block = 128 threads = 4 waves = the 4 column tiles of the 64-wide output.
// B (<=32KB) is staged in LDS once per block.
// f32 A 16x4 fragment layout: lane l (0..15) holds rows M=l, K=kb,kb+1 where
// kb = k0 + 2*(lane>>4). B 4x16 mirrors with N instead of M. C/D: VGPR v at
// lane l -> (M = v + 8*(l>>4), N = l&15).
// ---------------------------------------------------------------------------
template<int K, bool RELU, bool BIAS>
__global__ void gemm_wmma(const float* __restrict__ A, const float* __restrict__ B,
                          const float* __restrict__ bias, float* __restrict__ C) {
  __shared__ float Bs[K * 64];
  for (int i = threadIdx.x; i < K * 64; i += 128) Bs[i] = B[i];
  __syncthreads();
  const int lane = threadIdx.x & 31;
  const int wave = threadIdx.x >> 5;
  const int half = lane >> 4;
  const int l15  = lane & 15;
  const int m0   = blockIdx.x << 4;
  const int n0   = wave << 4;
  const float* Arow = A + (size_t)(m0 + l15) * K + half * 2;
  const float* Bcol = Bs + (size_t)(half * 2) * 64 + n0 + l15;
  v8f acc = {};
#pragma unroll 8
  for (int k0 = 0; k0 < K; k0 += 4) {
    v2f a, b;
    a.x = Arow[0]; a.y = Arow[1];
    b.x = Bcol[0]; b.y = Bcol[64];
    Arow += 4; Bcol += 4 * 64;
    acc = __builtin_amdgcn_wmma_f32_16x16x4_f32(false, a, false, b, (short)0, acc,
                                                false, false);
  }
  const float bv = BIAS ? bias[n0 + l15] : 0.0f;
#pragma unroll
  for (int v = 0; v < 8; ++v) {
    float val = acc[v] + bv;
    if (RELU) val = fmaxf(val, 0.0f);
    C[(size_t)(m0 + v + half * 8) * 64 + n0 + l15] = val;
  }
}

// ---------------------------------------------------------------------------
// 1D conv over the node axis as 5 shifted 64x64 WMMA GEMMs.
// X: [N,64] (node-major), W: [Cout=64][Cin=64][Ktap=5], Y: [N,64].
// Out-of-range taps are masked by multiply (no divergence -> EXEC all ones
// at every WMMA, per ISA restriction).
// ---------------------------------------------------------------------------
template<bool RELU>
__global__ void conv1d_wmma(const float* __restrict__ X, const float* __restrict__ W,
                            const float* __restrict__ bias, float* __restrict__ Y,
                            int N) {
  const int lane = threadIdx.x & 31;
  const int wave = threadIdx.x >> 5;
  const int half = lane >> 4;
  const int l15  = lane & 15;
  const int m0   = blockIdx.x << 4;
  const int n0   = wave << 4;
  const int co   = n0 + l15;
  v8f acc = {};
  for (int tap = 0; tap < 5; ++tap) {
    const int r    = m0 + l15 + tap - 2;
    const bool ok  = (r >= 0) && (r < N);
    const float msk = ok ? 1.0f : 0.0f;
    const float* Xr = X + (size_t)(ok ? r : 0) * 64 + half * 2;
    const float* Wc = W + (size_t)co * (64 * 5) + (half * 2) * 5 + tap;
#pragma unroll 8
    for (int k0 = 0; k0 < 64; k0 += 4) {
      v2f a, b;
      a.x = Xr[0] * msk; a.y = Xr[1] * msk;
      b.x = Wc[0];       b.y = Wc[5];
      Xr += 4; Wc += 4 * 5;
      acc = __builtin_amdgcn_wmma_f32_16x16x4_f32(false, a, false, b, (short)0, acc,
                                                  false, false);
    }
  }
  const float bv = bias[co];
#pragma unroll
  for (int v = 0; v < 8; ++v) {
    float val = acc[v] + bv;
    if (RELU) val = fmaxf(val, 0.0f);
    Y[(size_t)(m0 + v + half * 8) * 64 + n0 + l15] = val;
  }
}

// ------------------------------- GCN helpers -------------------------------
__global__ void deg_init_kernel(float* deg, int N) {
  int i = blockIdx.x * blockDim.x + threadIdx.x;
  if (i < N) deg[i] = 1.0f;  // self-loop
}

__global__ void deg_edge_kernel(const int* __restrict__ dst, float* deg, int E) {
  int e = blockIdx.x * blockDim.x + threadIdx.x;
  if (e < E) atomicAdd(&deg[dst[e]], 1.0f);
}

__global__ void dinv_kernel(const float* __restrict__ deg, float* dinv, int N) {
  int i = blockIdx.x * blockDim.x + threadIdx.x;
  if (i < N) dinv[i] = rsqrtf(deg[i]);  // deg >= 1 always
}

// out[i,f] = bias[f] + dinv[i]^2 * mm[i,f]   (self-loop term + bias)
__global__ void agg_init_kernel(const float* __restrict__ mm,
                                const float* __restrict__ dinv,
                                const float* __restrict__ bias,
                                float* __restrict__ out, int total) {
  int t = blockIdx.x * blockDim.x + threadIdx.x;
  if (t >= total) return;
  const int i = t >> 6, f = t & 63;
  const float di = dinv[i];
  out[t] = bias[f] + di * di * mm[t];
}

// 16 threads per edge, float4 each: out[dst] += dinv[s]*dinv[d] * mm[src]
__global__ void agg_edge_kernel(const float* __restrict__ mm,
                                const int* __restrict__ src,
                                const int* __restrict__ dst,
                                const float* __restrict__ dinv,
                                float* __restrict__ out, int E) {
  int t = blockIdx.x * blockDim.x + threadIdx.x;
  int e = t >> 4;
  if (e >= E) return;
  const int q = (t & 15) << 2;
  const int s = src[e], d = dst[e];
  const float norm = dinv[s] * dinv[d];
  const float* hs = mm + (size_t)s * 64 + q;
  float* od = out + (size_t)d * 64 + q;
#pragma unroll
  for (int j = 0; j < 4; ++j) atomicAdd(&od[j], norm * hs[j]);
}

__global__ void relu_kernel(float* x, int total) {
  int t = blockIdx.x * blockDim.x + threadIdx.x;
  if (t < total) x[t] = fmaxf(x[t], 0.0f);
}

// ------------------------------- pooling path ------------------------------
__global__ void score_kernel(const float* __restrict__ h1, const float* __restrict__ Wp,
                             const float* __restrict__ bp, float* __restrict__ score,
                             int N) {
  int i = blockIdx.x * blockDim.x + threadIdx.x;
  if (i >= N) return;
  float s = bp[0];
  const float* hr = h1 + (size_t)i * 64;
#pragma unroll 16
  for (int f = 0; f < 64; ++f) s += hr[f] * Wp[f];
  score[i] = s;
}

__global__ void sort_init_kernel(const float* __restrict__ score, float* keys,
                                 int* idxs, int N, int SZ) {
  int i = blockIdx.x * blockDim.x + threadIdx.x;
  if (i >= SZ) return;
  keys[i] = (i < N) ? score[i] : 3.402823e38f;  // pad sorts to the end (ascending)
  idxs[i] = i;
}

__global__ void bitonic_step_kernel(float* keys, int* idxs, int j, int k) {
  int i = blockIdx.x * blockDim.x + threadIdx.x;
  int ixj = i ^ j;
  if (ixj <= i) return;
  const bool up = ((i & k) == 0);
  float ki = keys[i], kj = keys[ixj];
  if (up ? (ki > kj) : (ki < kj)) {
    keys[i] = kj; keys[ixj] = ki;
    int t = idxs[i]; idxs[i] = idxs[ixj]; idxs[ixj] = t;
  }
}

// sortedX[r,f] = sorted_score[r] * h1[sorted_idx[r], f]
__global__ void gather_mul_kernel(const float* __restrict__ h1,
                                  const float* __restrict__ keys,
                                  const int* __restrict__ idxs,
                                  float* __restrict__ sortedX, int total) {
  int t = blockIdx.x * blockDim.x + threadIdx.x;
  if (t >= total) return;
  const int r = t >> 6, f = t & 63;
  sortedX[t] = keys[r] * h1[(size_t)idxs[r] * 64 + f];
}

// h2[sorted_idx[r], f] = conv_out[r, f]
__global__ void scatter_unsort_kernel(const float* __restrict__ convO,
                                      const int* __restrict__ idxs,
                                      float* __restrict__ h2, int total) {
  int t = blockIdx.x * blockDim.x + threadIdx.x;
  if (t >= total) return;
  const int r = t >> 6, f = t & 63;
  h2[(size_t)idxs[r] * 64 + f] = convO[t];
}

__global__ void concat_kernel(const float* __restrict__ h1, const float* __restrict__ h2,
                              float* __restrict__ hcat, int total) {
  int t = blockIdx.x * blockDim.x + threadIdx.x;
  if (t >= total) return;
  const int i = t >> 7, f = t & 127;
  hcat[t] = (f < 64) ? h1[(size_t)i * 64 + f] : h2[(size_t)i * 64 + (f - 64)];
}

// ---------------------------------------------------------------------------
static inline int ceil_div(int a, int b) { return (a + b - 1) / b; }

extern "C" void kernel_launch(void* const* d_in, const int* in_sizes, int n_in,
                              void* d_out, int out_size, void* d_ws, size_t ws_size,
                              hipStream_t stream) {
  const float* x       = (const float*)d_in[0];
  const int*   ei      = (const int*)  d_in[1];
  const float* W0      = (const float*)d_in[2];
  const float* b0      = (const float*)d_in[3];
  const float* W1      = (const float*)d_in[4];
  const float* b1      = (const float*)d_in[5];
  const float* W2      = (const float*)d_in[6];
  const float* b2      = (const float*)d_in[7];
  const float* Wp      = (const float*)d_in[8];
  const float* bp      = (const float*)d_in[9];
  const float* c1w     = (const float*)d_in[10];
  const float* c1b     = (const float*)d_in[11];
  const float* c2w     = (const float*)d_in[12];
  const float* c2b     = (const float*)d_in[13];
  const float* linw    = (const float*)d_in[14];
  const float* linb    = (const float*)d_in[15];
  float* out = (float*)d_out;

  const int N = in_sizes[0] / 128;   // 100000, multiple of 16
  const int E = in_sizes[1] / 2;     // 1600000
  const int* src = ei;
  const int* dst = ei + E;

  int SZ = 1;
  while (SZ < N) SZ <<= 1;           // 131072

  // workspace layout (floats)
  float* ws    = (float*)d_ws;
  float* deg   = ws;                         // N
  float* dinv  = deg + N;                    // N
  float* mm    = dinv + N;                   // N*64  (GEMM out / conv2 out)
  float* hA    = mm + (size_t)N * 64;        // N*64  (layer0 out / sortedX / h2)
  float* hB    = hA + (size_t)N * 64;        // N*64  (layer1 out / conv1 out)
  float* h1    = hB + (size_t)N * 64;        // N*64  (GCN result, kept)
  float* score = h1 + (size_t)N * 64;        // N
  float* keys  = score + N;                  // SZ
  int*   idxs  = (int*)(keys + SZ);          // SZ
  float* hcat  = (float*)(idxs + SZ);        // N*128

  const int BT = 256;
  const int NF = N * 64;             // elementwise size over [N,64]
  const int tilesM = N / 16;         // 6250 row tiles

  // ---- degree normalization -------------------------------------------------
  deg_init_kernel<<<ceil_div(N, BT), BT, 0, stream>>>(deg, N);
  deg_edge_kernel<<<ceil_div(E, BT), BT, 0, stream>>>(dst, deg, E);
  dinv_kernel<<<ceil_div(N, BT), BT, 0, stream>>>(deg, dinv, N);

  const int edgeThreads = E * 16;

  // ---- GCN layer 0: relu(agg(x @ W0) + b0) ---------------------------------
  gemm_wmma<128, false, false><<<tilesM, 128, 0, stream>>>(x, W0, nullptr, mm);
  agg_init_kernel<<<ceil_div(NF, BT), BT, 0, stream>>>(mm, dinv, b0, hA, NF);
  agg_edge_kernel<<<ceil_div(edgeThreads, BT), BT, 0, stream>>>(mm, src, dst, dinv, hA, E);
  relu_kernel<<<ceil_div(NF, BT), BT, 0, stream>>>(hA, NF);

  // ---- GCN layer 1 ----------------------------------------------------------
  gemm_wmma<64, false, false><<<tilesM, 128, 0, stream>>>(hA, W1, nullptr, mm);
  agg_init_kernel<<<ceil_div(NF, BT), BT, 0, stream>>>(mm, dinv, b1, hB, NF);
  agg_edge_kernel<<<ceil_div(edgeThreads, BT), BT, 0, stream>>>(mm, src, dst, dinv, hB, E);
  relu_kernel<<<ceil_div(NF, BT), BT, 0, stream>>>(hB, NF);

  // ---- GCN layer 2 (no relu) -> h1 -----------------------------------------
  gemm_wmma<64, false, false><<<tilesM, 128, 0, stream>>>(hB, W2, nullptr, mm);
  agg_init_kernel<<<ceil_div(NF, BT), BT, 0, stream>>>(mm, dinv, b2, h1, NF);
  agg_edge_kernel<<<ceil_div(edgeThreads, BT), BT, 0, stream>>>(mm, src, dst, dinv, h1, E);

  // ---- score + argsort (bitonic, ascending, padded with +inf) ---------------
  score_kernel<<<ceil_div(N, BT), BT, 0, stream>>>(h1, Wp, bp, score, N);
  sort_init_kernel<<<ceil_div(SZ, BT), BT, 0, stream>>>(score, keys, idxs, N, SZ);
  for (int k = 2; k <= SZ; k <<= 1)
    for (int j = k >> 1; j > 0; j >>= 1)
      bitonic_step_kernel<<<SZ / BT, BT, 0, stream>>>(keys, idxs, j, k);

  // ---- sorted_x = score * h1 gathered into rank order (-> hA) ---------------
  gather_mul_kernel<<<ceil_div(NF, BT), BT, 0, stream>>>(h1, keys, idxs, hA, NF);

  // ---- conv1 (relu) -> hB ; conv2 -> mm ------------------------------------
  conv1d_wmma<true ><<<tilesM, 128, 0, stream>>>(hA, c1w, c1b, hB, N);
  conv1d_wmma<false><<<tilesM, 128, 0, stream>>>(hB, c2w, c2b, mm, N);

  // ---- un-sort back to node order (-> hA as h2) -----------------------------
  scatter_unsort_kernel<<<ceil_div(NF, BT), BT, 0, stream>>>(mm, idxs, hA, NF);

  // ---- concat [h1 | h2] and final linear ------------------------------------
  concat_kernel<<<ceil_div(N * 128, BT), BT, 0, stream>>>(h1, hA, hcat, N * 128);
  gemm_wmma<128, false, true><<<tilesM, 128, 0, stream>>>(hcat, linw, linb, out);
}